// MeshConv_9964324127224
// MI455X (gfx1250) — compile-verified
//
#include <hip/hip_runtime.h>

typedef float v2f __attribute__((ext_vector_type(2)));
typedef float v8f __attribute__((ext_vector_type(8)));

#define F_DIM 16384
#define C_DIM 64
#define B_DIM 32
#define O_DIM 128
#define KDIM  128
#define NDIM  128
#define BLK_ROWS 128
#define ASTR 132           // padded A-tile stride (conflict-free frag reads)
#define WSTR 258           // words per K-pair row of W (even -> 8B aligned v2f)
#define NBLK 4096          // (B*F)/BLK_ROWS

// ---------------- Kernel 1: transpose x (B,C,F) -> xt (B*F, C) ----------------
__global__ __launch_bounds__(256) void k_transpose(const float* __restrict__ x,
                                                   float* __restrict__ xt) {
  __shared__ float tile[64][65];
  const int b   = blockIdx.x >> 8;          // 256 f-tiles per batch
  const int f0  = (blockIdx.x & 255) << 6;  // 64-wide f tile
  const int tid = threadIdx.x;
#pragma unroll
  for (int i = 0; i < 16; ++i) {
    int e = i * 256 + tid;
    int c = e >> 6, fl = e & 63;
    tile[c][fl] = x[((size_t)b * C_DIM + c) * F_DIM + (size_t)(f0 + fl)];
  }
  __syncthreads();
#pragma unroll
  for (int i = 0; i < 16; ++i) {
    int e = i * 256 + tid;
    int fl = e >> 6, c = e & 63;
    xt[((size_t)b * F_DIM + (size_t)(f0 + fl)) * C_DIM + c] = tile[c][fl];
  }
}

// ------- Kernel 2: gather + beta + fp32 WMMA GEMM + raw-y store + BN partials -------
__global__ __launch_bounds__(256) void k_gemm(const float* __restrict__ xt,
                                              const int* __restrict__ adj,
                                              const float* __restrict__ W,
                                              const float* __restrict__ bias,
                                              float* __restrict__ out,
                                              float* __restrict__ psum,
                                              float* __restrict__ psq) {
  extern __shared__ float smem[];
  float* Alds = smem;                         // BLK_ROWS * ASTR       (16896 f)
  float* Wlds = Alds + BLK_ROWS * ASTR;       // 64 kq * WSTR          (16512 f)
  float* Red  = Wlds + 64 * WSTR;             // 8 waves * 8 nt * 16 * 2 (2048 f)

  const int tid     = threadIdx.x;
  const int blk     = blockIdx.x;
  const int rowBase = blk * BLK_ROWS;         // global (b,f) row base
  const int b       = rowBase >> 14;          // F = 2^14; block stays in one b
  const int fBase   = rowBase & (F_DIM - 1);

  // Phase 0: stage W K-pair interleaved:  Wlds[kq*WSTR + n*2 + (k&1)] = Wcat[n][k]
  for (int e = tid; e < KDIM * NDIM; e += 256) {
    int n = e & 127, k = e >> 7;
    float w = (k < 64) ? W[n * 128 + k * 2] : W[n * 128 + (k - 64) * 2 + 1];
    Wlds[(k >> 1) * WSTR + n * 2 + (k & 1)] = w;
  }

  // Phase 1: gather neighbors, build A-tile rows = [target(64) | beta(64)]
  {
    const int c  = tid & 63;
    const int rg = tid >> 6;                  // 4 rows per pass
    const float* xb = xt + (size_t)b * F_DIM * C_DIM;
    for (int r0 = 0; r0 < BLK_ROWS; r0 += 4) {
      int r = r0 + rg;
      const int* a4 = adj + (size_t)(rowBase + r) * 4;
      int i0 = a4[0], i1 = a4[1], i2 = a4[2], i3 = a4[3];
      float t  = xb[(size_t)i0 * C_DIM + c];
      float n1 = xb[(size_t)i1 * C_DIM + c];
      float n2 = xb[(size_t)i2 * C_DIM + c];
      float n3 = xb[(size_t)i3 * C_DIM + c];
      Alds[r * ASTR + c]      = t;
      Alds[r * ASTR + 64 + c] = fabsf(t - n1) + fabsf(t - n2) + fabsf(t - n3);
    }
  }
  __syncthreads();

  // Phase 2: WMMA.  wave w owns rows [w*16, w*16+16) x all 128 channels.
  const int wave = tid >> 5;
  const int lane = tid & 31;
  const int lh   = lane & 15;
  const bool hi  = lane >= 16;
  const int arow = wave * 16 + lh;
  const int koff = hi ? 2 : 0;                // A/B frag K sub-offset per half-wave

  v8f acc[8];
#pragma unroll
  for (int nt = 0; nt < 8; ++nt) {
    float bv = bias[nt * 16 + lh];            // C init = bias[n], broadcast over M
#pragma unroll
    for (int i = 0; i < 8; ++i) acc[nt][i] = bv;
  }

  // per-lane base into the pair-interleaved W tile
  const float* wbase = Wlds + (hi ? WSTR : 0) + lh * 2;

  for (int kt = 0; kt < 32; ++kt) {
    int k0 = kt * 4 + koff;
    v2f af = *(const v2f*)&Alds[arow * ASTR + k0];        // A[arow][k0], [k0+1]
    const float* wk = wbase + (kt * 2) * WSTR;
#pragma unroll
    for (int nt = 0; nt < 8; ++nt) {
      v2f bf = *(const v2f*)&wk[nt * 32];                 // {B[k0][n], B[k0+1][n]}
      acc[nt] = __builtin_amdgcn_wmma_f32_16x16x4_f32(
          false, af, false, bf, (short)0, acc[nt], false, false);
    }
  }

  // Phase 3: store raw y + per-channel partial sums (deterministic reduce)
  const int f0 = fBase + wave * 16 + (hi ? 8 : 0);   // acc[nt][i] -> f = f0 + i
#pragma unroll
  for (int nt = 0; nt < 8; ++nt) {
    int n = nt * 16 + lh;
    float* op = out + ((size_t)(b * O_DIM + n) << 14) + (size_t)f0;
    v8f a = acc[nt];
    float4 vlo = make_float4(a[0], a[1], a[2], a[3]);
    float4 vhi = make_float4(a[4], a[5], a[6], a[7]);
    *(float4*)(op)     = vlo;
    *(float4*)(op + 4) = vhi;
    float s = 0.f, s2 = 0.f;
#pragma unroll
    for (int i = 0; i < 8; ++i) { float v = a[i]; s += v; s2 += v * v; }
    s  += __shfl_xor(s, 16);                  // combine M rows 0-7 with 8-15
    s2 += __shfl_xor(s2, 16);
    if (!hi) {
      Red[((wave * 8 + nt) * 16 + lh) * 2 + 0] = s;
      Red[((wave * 8 + nt) * 16 + lh) * 2 + 1] = s2;
    }
  }
  __syncthreads();
  if (tid < 128) {                             // one thread per channel
    int ntt = tid >> 4, j = tid & 15;
    float S = 0.f, S2 = 0.f;
#pragma unroll
    for (int w = 0; w < 8; ++w) {
      S  += Red[((w * 8 + ntt) * 16 + j) * 2 + 0];
      S2 += Red[((w * 8 + ntt) * 16 + j) * 2 + 1];
    }
    psum[(size_t)blk * 128 + tid] = S;
    psq [(size_t)blk * 128 + tid] = S2;
  }
}

// ---------------- Kernel 3: reduce partials -> per-channel scale/shift ----------------
__global__ __launch_bounds__(256) void k_bnstats(const float* __restrict__ psum,
                                                 const float* __restrict__ psq,
                                                 const float* __restrict__ gamma,
                                                 const float* __restrict__ beta,
                                                 float* __restrict__ sc,
                                                 float* __restrict__ sh) {
  __shared__ float r1[256], r2[256];
  const int n = blockIdx.x, tid = threadIdx.x;
  float s = 0.f, s2 = 0.f;
  for (int i = tid; i < NBLK; i += 256) {
    s  += psum[(size_t)i * 128 + n];
    s2 += psq [(size_t)i * 128 + n];
  }
  r1[tid] = s; r2[tid] = s2;
  __syncthreads();
  for (int off = 128; off > 0; off >>= 1) {
    if (tid < off) { r1[tid] += r1[tid + off]; r2[tid] += r2[tid + off]; }
    __syncthreads();
  }
  if (tid == 0) {
    const float inv  = 1.0f / (float)(B_DIM * F_DIM);
    float mean = r1[0] * inv;
    float var  = r2[0] * inv - mean * mean;
    float g    = gamma[n] * rsqrtf(var + 1e-5f);
    sc[n] = g;
    sh[n] = beta[n] - mean * g;
  }
}

// ---------------- Kernel 4: in-place normalize + ReLU ----------------
__global__ __launch_bounds__(256) void k_bnapply(float* __restrict__ out,
                                                 const float* __restrict__ sc,
                                                 const float* __restrict__ sh) {
  size_t q = (size_t)blockIdx.x * 256 + threadIdx.x;   // float4 index
  int n = (int)(q >> 12) & 127;                         // 4096 float4 per channel row
  float g = sc[n], s = sh[n];
  float4 v = ((float4*)out)[q];
  v.x = fmaxf(v.x * g + s, 0.f);
  v.y = fmaxf(v.y * g + s, 0.f);
  v.z = fmaxf(v.z * g + s, 0.f);
  v.w = fmaxf(v.w * g + s, 0.f);
  ((float4*)out)[q] = v;
}

extern "C" void kernel_launch(void* const* d_in, const int* in_sizes, int n_in,
                              void* d_out, int out_size, void* d_ws, size_t ws_size,
                              hipStream_t stream) {
  const float* x     = (const float*)d_in[0];
  const int*   adj   = (const int*)d_in[1];
  const float* W     = (const float*)d_in[2];
  const float* bias  = (const float*)d_in[3];
  const float* gamma = (const float*)d_in[4];
  const float* beta  = (const float*)d_in[5];
  float* out = (float*)d_out;

  char* ws = (char*)d_ws;
  float* xt   = (float*)ws;                                    // 128 MB
  float* psum = (float*)(ws + (size_t)B_DIM * F_DIM * C_DIM * 4);
  float* psq  = psum + (size_t)NBLK * 128;                     // 2 MB each
  float* sc   = psq + (size_t)NBLK * 128;
  float* sh   = sc + 128;

  // 1) transpose
  k_transpose<<<B_DIM * (F_DIM / 64), 256, 0, stream>>>(x, xt);

  // 2) fused gather + WMMA GEMM (dynamic LDS)
  const int smemBytes = (BLK_ROWS * ASTR + 64 * WSTR + 2048) * 4;
  hipFuncSetAttribute(reinterpret_cast<const void*>(k_gemm),
                      hipFuncAttributeMaxDynamicSharedMemorySize, smemBytes);
  k_gemm<<<NBLK, 256, smemBytes, stream>>>(xt, adj, W, bias, out, psum, psq);

  // 3) BN statistics
  k_bnstats<<<O_DIM, 256, 0, stream>>>(psum, psq, gamma, beta, sc, sh);

  // 4) normalize + ReLU
  k_bnapply<<<(B_DIM * O_DIM * F_DIM / 4) / 256, 256, 0, stream>>>(out, sc, sh);
}